// MetaGIN_60009283059906
// MI455X (gfx1250) — compile-verified
//
#include <hip/hip_runtime.h>
#include <hip/hip_bf16.h>
#include <stdint.h>

// MetaGIN layer for MI455X (gfx1250, wave32).
// fp32 end-to-end; dense + grouped GEMMs on V_WMMA_F32_16X16X4_F32.
// Weight matrices staged to LDS with gfx1250 async global->LDS copies.

typedef __attribute__((ext_vector_type(2))) float v2f;
typedef __attribute__((ext_vector_type(8))) float v8f;

#define WIDTH  128
#define NHEAD  8
#define GSIZE  16
#define GN_EPS 1e-5f
#define WPAD   132   // LDS stride for weights: 528B = 16B-aligned rows, 132%64=4 -> conflict-free
#define LPAD   130   // LDS stride for edge tiles: 8B-aligned rows, 130%64=2 -> conflict-free
#define ETILE  32    // edges per block

static __device__ __forceinline__ v8f wmma_f32_16x16x4(v2f a, v2f b, v8f c) {
  // 8 args: (neg_a, A, neg_b, B, c_mod, C, reuse_a, reuse_b)
  return __builtin_amdgcn_wmma_f32_16x16x4_f32(false, a, false, b, (short)0, c,
                                               false, false);
}

// Async copy of one 16B chunk: global -> LDS (VDST = LDS byte offset VGPR,
// VADDR = 64-bit global address). Tracked by ASYNCcnt.
static __device__ __forceinline__ void async_g2l_b128(unsigned lds_off,
                                                      const float* gaddr) {
  asm volatile("global_load_async_to_lds_b128 %0, %1, off"
               :: "v"(lds_off), "v"(gaddr) : "memory");
}
static __device__ __forceinline__ void wait_async0() {
  asm volatile("s_wait_asynccnt 0" ::: "memory");
}

// Stage a 128x128 f32 weight matrix (row-major) into LDS with WPAD stride.
// 4096 16B chunks over 256 threads = 16 chunks/thread.
static __device__ __forceinline__ void stage_weights(float (*s_w)[WPAD],
                                                     const float* __restrict__ W,
                                                     int t) {
#pragma unroll
  for (int c = 0; c < 16; ++c) {
    const int idx = c * 256 + t;
    const int row = idx >> 5;        // 0..127
    const int seg = idx & 31;        // 0..31 (16B chunks of a 512B row)
    const unsigned l = (unsigned)(uintptr_t)(&s_w[row][seg * 4]);
    async_g2l_b128(l, W + (size_t)row * WIDTH + seg * 4);
  }
  wait_async0();
}

// ---------------------------------------------------------------- zero fill
__global__ __launch_bounds__(256) void zero_kernel(float* __restrict__ p, long n) {
  long i = (long)blockIdx.x * blockDim.x + threadIdx.x;
  long stride = (long)gridDim.x * blockDim.x;
  for (; i < n; i += stride) p[i] = 0.0f;
}

// ------------------------------------------------- node projections h0, h1
// Block (256 thr = 8 waves) stages one weight matrix to LDS, then each wave
// computes a 16-row x 128-col strip: A from global (b64/lane), B from LDS.
// WMMA f32 16x16x4 layouts (ISA 7.12.2):
//   A: lanes 0-15 M=lane K=k0+{0,1}; lanes 16-31 K=k0+{2,3}
//   B: lanes 0-15 N=lane K=k0+{0,1}; lanes 16-31 K=k0+{2,3}
//   C/D: VGPR r -> M=r / r+8 (lane halves), N=lane&15
__global__ __launch_bounds__(256) void node_proj_kernel(
    const float* __restrict__ x,
    const float* __restrict__ w0, const float* __restrict__ b0,
    const float* __restrict__ w1, const float* __restrict__ b1,
    float* __restrict__ h0, float* __restrict__ h1, int n) {
  __shared__ float s_w[WIDTH][WPAD];   // ~66 KB of the 320 KB WGP LDS

  const int t   = (int)threadIdx.x;
  const int mat = (int)blockIdx.x & 1;
  const int grp = (int)blockIdx.x >> 1;

  const float* __restrict__ W    = mat ? w1 : w0;
  const float* __restrict__ bias = mat ? b1 : b0;
  float* __restrict__ H          = mat ? h1 : h0;

  stage_weights(s_w, W, t);
  __syncthreads();                     // all waves' async copies landed

  const int wavei = t >> 5;
  const int lane  = t & 31;
  const int rt    = grp * 8 + wavei;
  if (rt * 16 >= n) return;            // after the only barrier: safe

  int row = rt * 16 + (lane & 15);
  if (row >= n) row = n - 1;           // clamp keeps EXEC full for WMMA
  const int khalf = (lane >> 4) << 1;  // 0 or 2
  const int ncol  = lane & 15;

  v8f acc[8] = {};
  for (int k0 = 0; k0 < WIDTH; k0 += 4) {
    const v2f a = *(const v2f*)(x + (size_t)row * WIDTH + k0 + khalf);
#pragma unroll
    for (int tt = 0; tt < 8; ++tt) {
      const int col = tt * 16 + ncol;
      v2f b;
      b.x = s_w[col][k0 + khalf];
      b.y = s_w[col][k0 + khalf + 1];
      acc[tt] = wmma_f32_16x16x4(a, b, acc[tt]);
    }
  }

  const int radd = (lane >> 4) << 3;   // 0 or 8
#pragma unroll
  for (int tt = 0; tt < 8; ++tt) {
    const int col  = tt * 16 + ncol;
    const float bv = bias[col];
#pragma unroll
    for (int r = 0; r < 8; ++r) {
      const int rr = rt * 16 + radd + r;
      if (rr < n) H[(size_t)rr * WIDTH + col] = acc[tt][r] + bv;
    }
  }
}

// --------------------------------------------------------------- edge kernel
// 32 edges / block (256 threads = 8 waves).
// Phase 0: gather h0[src]+h1[dst] -> LDS.  Phase 1: GroupNorm stats
// (one thread per (edge,group)).  Phase 2: normalize + EmbeddingBag(mean,
// padding_idx=0), one thread per (edge,group).  Phase 3: wave g computes
// group g's gate/value 16x16 GEMMs via WMMA for two 16-edge tiles (B kept
// in registers), then relu(gate)*value scatter-added with f32 atomics.
__global__ __launch_bounds__(256) void edge_kernel(
    const float* __restrict__ h0, const float* __restrict__ h1,
    const int* __restrict__ edge_idx, const int* __restrict__ edge_attr,
    const float* __restrict__ emb, const float* __restrict__ gate_w,
    const float* __restrict__ value_w, float* __restrict__ agg, int e) {
  __shared__ float s_xx[ETILE][LPAD];   // raw, then normalized xx
  __shared__ float s_xb[ETILE][LPAD];   // normalized xx + bag
  __shared__ float s_m[ETILE][NHEAD];
  __shared__ float s_inv[ETILE][NHEAD];
  __shared__ int   s_dst[ETILE];

  const int t     = (int)threadIdx.x;
  const int ebase = (int)blockIdx.x * ETILE;

  { // ---- phase 0: gather (16 floats / thread / source, float4 loads)
    const int el = t >> 3;             // 0..31
    const int cb = (t & 7) * 16;       // 0..112
    int ge = ebase + el;
    if (ge >= e) ge = e - 1;
    const int src = edge_idx[ge];
    const int dst = edge_idx[e + ge];
    if ((t & 7) == 0) s_dst[el] = dst;
    const float4* p0 = (const float4*)(h0 + (size_t)src * WIDTH + cb);
    const float4* p1 = (const float4*)(h1 + (size_t)dst * WIDTH + cb);
#pragma unroll
    for (int q = 0; q < 4; ++q) {
      const float4 a = p0[q];
      const float4 b = p1[q];
      s_xx[el][cb + 4 * q + 0] = a.x + b.x;
      s_xx[el][cb + 4 * q + 1] = a.y + b.y;
      s_xx[el][cb + 4 * q + 2] = a.z + b.z;
      s_xx[el][cb + 4 * q + 3] = a.w + b.w;
    }
  }
  __syncthreads();

  { // ---- phase 1: per-(edge,group) mean / rsqrt(var+eps); 256 pairs, 256 thr
    const int el = t >> 3;
    const int g  = t & 7;
    float s = 0.0f, s2 = 0.0f;
#pragma unroll
    for (int j = 0; j < GSIZE; ++j) {
      const float v = s_xx[el][g * GSIZE + j];
      s += v; s2 += v * v;
    }
    const float m   = s * (1.0f / GSIZE);
    const float var = s2 * (1.0f / GSIZE) - m * m;
    s_m[el][g]   = m;
    s_inv[el][g] = rsqrtf(var + GN_EPS);
  }
  __syncthreads();

  { // ---- phase 2: normalize + embedding bag (one thread per (edge,group))
    const int el = t >> 3;
    const int g  = t & 7;
    int ge = ebase + el;
    if (ge >= e) ge = e - 1;
    const int a0 = edge_attr[ge * 3 + 0];
    const int a1 = edge_attr[ge * 3 + 1];
    const int a2 = edge_attr[ge * 3 + 2];
    const int nz = (a0 != 0) + (a1 != 0) + (a2 != 0);
    const float rcnt = 1.0f / (float)(nz > 0 ? nz : 1);
    const float m   = s_m[el][g];
    const float inv = s_inv[el][g];
#pragma unroll
    for (int j = 0; j < GSIZE; ++j) {
      const int c = g * GSIZE + j;
      float bg = 0.0f;
      if (a0) bg += emb[(size_t)a0 * WIDTH + c];
      if (a1) bg += emb[(size_t)a1 * WIDTH + c];
      if (a2) bg += emb[(size_t)a2 * WIDTH + c];
      bg *= rcnt;
      const float xn = (s_xx[el][c] - m) * inv;
      s_xx[el][c] = xn;
      s_xb[el][c] = xn + bg;
    }
  }
  __syncthreads();

  { // ---- phase 3: grouped conv (WMMA) + scatter; wave g owns head g
    const int g     = t >> 5;
    const int lane  = t & 31;
    const int mrow  = lane & 15;
    const int nout  = lane & 15;
    const int khalf = (lane >> 4) << 1;

    // B fragments (gate/value weights of this group) live in registers and
    // are reused across both edge tiles.
    v2f bg[4], bv[4];
#pragma unroll
    for (int kk4 = 0; kk4 < 4; ++kk4) {
      const int k = kk4 * 4 + khalf;
      bg[kk4] = *(const v2f*)(gate_w  + (size_t)(g * GSIZE + nout) * GSIZE + k);
      bv[kk4] = *(const v2f*)(value_w + (size_t)(g * GSIZE + nout) * GSIZE + k);
    }

#pragma unroll
    for (int tile = 0; tile < 2; ++tile) {
      const int erow = tile * 16 + mrow;
      v8f gacc = {}, vacc = {};
#pragma unroll
      for (int kk4 = 0; kk4 < 4; ++kk4) {
        const int k = kk4 * 4 + khalf;
        v2f ag, av;
        ag.x = s_xb[erow][g * GSIZE + k];
        ag.y = s_xb[erow][g * GSIZE + k + 1];
        av.x = s_xx[erow][g * GSIZE + k];
        av.y = s_xx[erow][g * GSIZE + k + 1];
        gacc = wmma_f32_16x16x4(ag, bg[kk4], gacc);
        vacc = wmma_f32_16x16x4(av, bv[kk4], vacc);
      }
      const int radd = (lane >> 4) << 3;
#pragma unroll
      for (int r = 0; r < 8; ++r) {
        const int el = tile * 16 + radd + r;
        if (ebase + el < e) {
          const float y = fmaxf(gacc[r], 0.0f) * vacc[r];
          atomicAdd(agg + (size_t)s_dst[el] * WIDTH + g * GSIZE + nout, y);
        }
      }
    }
  }
}

// --------------------------------- post projection + degree^param scaling
__global__ __launch_bounds__(256) void post_kernel(
    const float* __restrict__ agg, const float* __restrict__ W,
    const float* __restrict__ bias, const float* __restrict__ deg,
    const float* __restrict__ dpar, float* __restrict__ out, int n) {
  __shared__ float s_w[WIDTH][WPAD];

  const int t   = (int)threadIdx.x;
  const int grp = (int)blockIdx.x;

  stage_weights(s_w, W, t);
  __syncthreads();

  const int wavei = t >> 5;
  const int lane  = t & 31;
  const int rt    = grp * 8 + wavei;
  if (rt * 16 >= n) return;

  int row = rt * 16 + (lane & 15);
  if (row >= n) row = n - 1;
  const int khalf = (lane >> 4) << 1;
  const int ncol  = lane & 15;

  v8f acc[8] = {};
  for (int k0 = 0; k0 < WIDTH; k0 += 4) {
    const v2f a = *(const v2f*)(agg + (size_t)row * WIDTH + k0 + khalf);
#pragma unroll
    for (int tt = 0; tt < 8; ++tt) {
      const int col = tt * 16 + ncol;
      v2f b;
      b.x = s_w[col][k0 + khalf];
      b.y = s_w[col][k0 + khalf + 1];
      acc[tt] = wmma_f32_16x16x4(a, b, acc[tt]);
    }
  }

  const int radd = (lane >> 4) << 3;
#pragma unroll
  for (int tt = 0; tt < 8; ++tt) {
    const int col  = tt * 16 + ncol;
    const float bv = bias[col];
    const float dp = dpar[col];
#pragma unroll
    for (int r = 0; r < 8; ++r) {
      const int rr = rt * 16 + radd + r;
      if (rr < n)
        out[(size_t)rr * WIDTH + col] = powf(deg[rr], dp) * (acc[tt][r] + bv);
    }
  }
}

// -------------------------------------------------------------------- launch
extern "C" void kernel_launch(void* const* d_in, const int* in_sizes, int n_in,
                              void* d_out, int out_size, void* d_ws, size_t ws_size,
                              hipStream_t stream) {
  (void)n_in; (void)out_size; (void)ws_size;
  const float* x       = (const float*)d_in[0];
  const float* deg     = (const float*)d_in[1];
  const int*   eidx    = (const int*)d_in[2];
  const int*   eattr   = (const int*)d_in[3];
  const float* pre0_w  = (const float*)d_in[4];
  const float* pre0_b  = (const float*)d_in[5];
  const float* pre1_w  = (const float*)d_in[6];
  const float* pre1_b  = (const float*)d_in[7];
  const float* emb     = (const float*)d_in[8];
  const float* gate_w  = (const float*)d_in[9];
  const float* value_w = (const float*)d_in[10];
  const float* post_w  = (const float*)d_in[11];
  const float* post_b  = (const float*)d_in[12];
  const float* dpar    = (const float*)d_in[13];
  float* out = (float*)d_out;

  const int n = in_sizes[0] / WIDTH;   // 200000
  const int e = in_sizes[2] / 2;       // 1000000
  const size_t n128 = (size_t)n * WIDTH;

  float* h0  = (float*)d_ws;           // [N,128]
  float* h1  = h0 + n128;              // [N,128]
  float* agg = h1 + n128;              // [N,128] atomic accumulator

  // agg must be zero every call (ws is poisoned once, atomics accumulate)
  zero_kernel<<<1024, 256, 0, stream>>>(agg, (long)n128);

  {
    const int row_tiles  = (n + 15) / 16;
    const int tile_grps  = (row_tiles + 7) / 8;      // 8 row tiles / block
    const int blocks     = tile_grps * 2;            // x {pre0, pre1}
    node_proj_kernel<<<blocks, 256, 0, stream>>>(x, pre0_w, pre0_b,
                                                 pre1_w, pre1_b, h0, h1, n);
  }
  {
    const int blocks = (e + ETILE - 1) / ETILE;
    edge_kernel<<<blocks, 256, 0, stream>>>(h0, h1, eidx, eattr, emb,
                                            gate_w, value_w, agg, e);
  }
  {
    const int row_tiles = (n + 15) / 16;
    const int blocks    = (row_tiles + 7) / 8;
    post_kernel<<<blocks, 256, 0, stream>>>(agg, post_w, post_b, deg, dpar,
                                            out, n);
  }
}